// NaiveBayes_43361989820916
// MI455X (gfx1250) — compile-verified
//
#include <hip/hip_runtime.h>
#include <hip/hip_bf16.h>

// Naive Bayes posterior: x[N, 40] (20 feats + 20 mask) -> out[N, 8].
// HBM-bound: 192 B/row, ~384 MB total -> ~16.5 us floor at 23.3 TB/s.
// gfx1250 path: GLOBAL_LOAD_ASYNC_TO_LDS_B64 staging (ASYNCcnt) with a
// 168-byte-stride LDS layout that is provably bank-conflict-free for wave32
// (42-dword stride: 42*dt % 64 == 0 only for dt % 32 == 0), then 15 native
// v_exp_f32 Gaussian factors per row (the only distinct (col,mean) pairs).

#define NF     20          // features
#define NCOLS  40          // feats + mask
#define NC     8           // classes
#define RPB    256         // rows per block == threads per block (8 waves)
#define RSTRIDE 42         // LDS row stride in dwords (168 B)

#define INV_STD   3.3333333333333335f          // 1/0.3
#define INV_NORM  1.3298076013381093f          // 1/(0.3*sqrt(2*pi))

// Byte offset of a __shared__ object within the wave's LDS allocation.
__device__ __forceinline__ unsigned lds_off(const void* p) {
  return (unsigned)(unsigned long long)
      (__attribute__((address_space(3))) const void*)p;
}

// Stage rowsHere*160 bytes into LDS at 168 B row stride using async b64 copies.
// Chunk h = i*RPB + t covers 8 bytes; row = h/20, pair = h%20.
template <bool FULL>
__device__ __forceinline__ void stage_tile(const float* gbase, unsigned ldsBase,
                                           int t, int chunksHere) {
  int r  = t / 20;
  int k2 = t - r * 20;
  unsigned goff = (unsigned)(t * 8);
#pragma unroll
  for (int i = 0; i < 20; ++i) {
    if (FULL || (i * RPB + t) < chunksHere) {
      unsigned loff = ldsBase + (unsigned)(r * (RSTRIDE * 4) + k2 * 8);
      asm volatile("global_load_async_to_lds_b64 %0, %1, %2"
                   :: "v"(loff), "v"(goff), "s"(gbase)
                   : "memory");
    }
    // advance by 256 chunks: r += 12, k2 += 16, carry if k2 >= 20
    k2 += 16; r += 12;
    if (k2 >= 20) { k2 -= 20; r += 1; }
    goff += (unsigned)(RPB * 8);
  }
}

__global__ void __launch_bounds__(RPB)
nb_kernel(const float* __restrict__ x, float* __restrict__ out, int nrows) {
  __shared__ float tile[RSTRIDE * RPB];   // 43008 B, 168 B per row

  const int t        = threadIdx.x;
  const int rowBase  = blockIdx.x * RPB;
  const int rowsHere = min(RPB, nrows - rowBase);

  const float*   gbase   = x + (long long)rowBase * NCOLS;
  const unsigned ldsBase = lds_off(tile);

  if (rowsHere == RPB) {
    stage_tile<true>(gbase, ldsBase, t, RPB * 20);
  } else {
    stage_tile<false>(gbase, ldsBase, t, rowsHere * 20);
  }
  asm volatile("s_wait_asynccnt 0x0" ::: "memory");
  __syncthreads();

  // ---- Compute: one row per thread, conflict-free ds_load_b64 reads ----
  if (t < rowsHere) {
    const float* row = &tile[t * RSTRIDE];
    float f[NF], m[NF];
#pragma unroll
    for (int k = 0; k < NF; ++k) f[k] = row[k];
#pragma unroll
    for (int k = 0; k < NF; ++k) m[k] = row[NF + k];

    // bit k set <=> masked feature k falls outside [0,1): zeroes the product
    unsigned bad = 0u;
#pragma unroll
    for (int k = 0; k < NF; ++k) {
      bool inb = (f[k] >= 0.0f) && (f[k] < 1.0f);
      if ((m[k] > 0.0f) && !inb) bad |= (1u << k);
    }

    // Masked Gaussian factor for (column c, mean): only 15 distinct pairs used.
    auto mpdf = [&](int c, float mean) -> float {
      float z = (f[c] - mean) * INV_STD;
      float e = INV_NORM * __expf(-0.5f * z * z);
      return (m[c] > 0.0f) ? e : 1.0f;
    };
    float q0[7];   // mean 0: columns 0..6
#pragma unroll
    for (int c = 0; c < 7; ++c)  q0[c] = mpdf(c, 0.0f);
    float q1[10];  // mean 1: columns 1..9
#pragma unroll
    for (int c = 1; c < 10; ++c) q1[c] = mpdf(c, 1.0f);

    // Indicator over the 17 non-window columns, times prior 1/8.
    float ind[NC];
#pragma unroll
    for (int y = 0; y < NC; ++y)
      ind[y] = ((bad & ~(7u << y)) == 0u) ? 0.125f : 0.0f;

    float o[NC];
    o[0] = ind[0] * q0[0] * q0[1] * q0[2];
    o[1] = ind[1] * q1[1] * q0[2] * q0[3];
    o[2] = ind[2] * q0[2] * q1[3] * q0[4];
    o[3] = ind[3] * q1[3] * q1[4] * q0[5];
    o[4] = ind[4] * q0[4] * q0[5] * q1[6];
    o[5] = ind[5] * q1[5] * q0[6] * q1[7];
    o[6] = ind[6] * q0[6] * q1[7] * q1[8];
    o[7] = ind[7] * q1[7] * q1[8] * q1[9];

    float sum = ((o[0] + o[1]) + (o[2] + o[3])) +
                ((o[4] + o[5]) + (o[6] + o[7]));
    const float inv = 1.0f / fmaxf(sum, 1e-8f);

    float4* op = (float4*)(out + (long long)(rowBase + t) * NC);
    op[0] = make_float4(o[0] * inv, o[1] * inv, o[2] * inv, o[3] * inv);
    op[1] = make_float4(o[4] * inv, o[5] * inv, o[6] * inv, o[7] * inv);
  }
}

extern "C" void kernel_launch(void* const* d_in, const int* in_sizes, int n_in,
                              void* d_out, int out_size, void* d_ws, size_t ws_size,
                              hipStream_t stream) {
  const float* x   = (const float*)d_in[0];
  float*       out = (float*)d_out;
  const int nrows  = in_sizes[0] / NCOLS;
  const int blocks = (nrows + RPB - 1) / RPB;
  nb_kernel<<<blocks, RPB, 0, stream>>>(x, out, nrows);
}